// DeepConvGraphEncoderPre_36618891166131
// MI455X (gfx1250) — compile-verified
//
#include <hip/hip_runtime.h>
#include <cstdint>

typedef __attribute__((ext_vector_type(2))) float v2f;
typedef __attribute__((ext_vector_type(8))) float v8f;

// ---------------------------------------------------------------------------
// Graph preprocessing: symmetric GCN norm + deterministic CSR (dst-major).
// One block of 256 threads (N == 256 nodes).
// ---------------------------------------------------------------------------
__global__ void k_build_graph(const int* __restrict__ ei, int E,
                              int* __restrict__ off, int* __restrict__ srcout,
                              float* __restrict__ wout, float* __restrict__ dinv)
{
    __shared__ int deg[256];
    __shared__ int offs[257];
    const int t = threadIdx.x;            // node id
    deg[t] = 0;
    __syncthreads();
    for (int e = t; e < E; e += 256) atomicAdd(&deg[ei[E + e]], 1);  // ei[1][e] = dst
    __syncthreads();
    dinv[t] = rsqrtf((float)(deg[t] + 1));                           // +1: self loop
    if (t == 0) {
        int acc = 0;
        for (int v = 0; v < 256; ++v) { offs[v] = acc; acc += deg[v]; }
        offs[256] = acc;
    }
    __syncthreads();                      // also makes dinv[] visible in-block
    off[t] = offs[t];
    if (t == 0) off[256] = offs[256];
    // deterministic fill: thread t owns dst node t, edge order preserved
    int pos = offs[t];
    const float dv = dinv[t];
    for (int e = 0; e < E; ++e) {
        if (ei[E + e] == t) {
            const int s = ei[e];
            srcout[pos] = s;
            wout[pos]   = dinv[s] * dv;
            ++pos;
        }
    }
}

// ---------------------------------------------------------------------------
// Repack weight matrix into WMMA pair-interleaved layout:
//   Bp[kk*N + n] = ( B[2kk][n], B[2kk+1][n] )        (logical B is [K,N])
// trans==1: source stored [N,K] row-major (logical B = source^T).
// One v2f per thread; grid covers (K/2)*N elements.
// ---------------------------------------------------------------------------
__global__ void k_pack_B(const float* __restrict__ B, v2f* __restrict__ Bp,
                         int K, int N, int trans)
{
    const int idx = blockIdx.x * blockDim.x + threadIdx.x;
    const int tot = (K >> 1) * N;
    if (idx >= tot) return;
    const int kk = idx / N, n = idx % N;
    v2f p;
    if (trans) {
        p = *(const v2f*)(B + (size_t)n * K + 2 * kk);       // contiguous pair
    } else {
        p.x = B[(size_t)(2 * kk)     * N + n];
        p.y = B[(size_t)(2 * kk + 1) * N + n];
    }
    Bp[idx] = p;
}

// ---------------------------------------------------------------------------
// fp32 WMMA GEMM: C[M,N] = A[M,K] * B (+bias,+bias2), B pre-packed (k-pairs).
// One wave owns a 16 x (16*NT) strip: A fragment loaded once per K-step
// (global_load_b64) and reused across NT column tiles; each B fragment is a
// single global_load_b64 from the packed layout.
// Requires: K % 4 == 0, N % (16*NT) == 0. Ragged M handled by a wave-uniform
// versioned loop so the main path stays branch-free.
// ---------------------------------------------------------------------------
template<int NT>
__global__ void k_wmma_gemm(const float* __restrict__ A, const v2f* __restrict__ Bp,
                            const float* __restrict__ bias, const float* __restrict__ bias2,
                            float* __restrict__ C, int M, int N, int K)
{
    const int lane = threadIdx.x & 31;
    const int wave = threadIdx.x >> 5;
    const int grp  = N / (16 * NT);                 // column-tile groups
    const int tiles_m = (M + 15) >> 4;
    const int gid = blockIdx.x * (blockDim.x >> 5) + wave;
    if (gid >= tiles_m * grp) return;               // wave-uniform: EXEC stays all-1s
    const int tm  = gid / grp;
    const int tn0 = (gid % grp) * NT;

    const int ml   = lane & 15;
    const int half = lane >> 4;                     // K-pair select (A/B), M-half (C/D)
    const int arow = tm * 16 + ml;

    const v8f vzero = {0.f,0.f,0.f,0.f,0.f,0.f,0.f,0.f};
    v8f acc[NT];
#pragma unroll
    for (int j = 0; j < NT; ++j) acc[j] = vzero;

    if (tm * 16 + 16 <= M) {
        // ---- full tile: branch-free hot loop
        const float* ap = A + (size_t)arow * K + 2 * half;
        for (int k = 0; k < K; k += 4) {
            const v2f a = *(const v2f*)(ap + k);                       // b64
            const v2f* bp = Bp + ((size_t)((k >> 1) + half)) * N;
#pragma unroll
            for (int j = 0; j < NT; ++j) {
                const v2f b = bp[(tn0 + j) * 16 + ml];                 // b64
                acc[j] = __builtin_amdgcn_wmma_f32_16x16x4_f32(
                    false, a, false, b, (short)0, acc[j], false, false);
            }
        }
    } else {
        // ---- ragged final row tile (only the M=8 output projection)
        const bool rowOK = arow < M;
        for (int k = 0; k < K; k += 4) {
            v2f a;
            if (rowOK) a = *(const v2f*)(A + (size_t)arow * K + k + 2 * half);
            else { a.x = 0.f; a.y = 0.f; }
            const v2f* bp = Bp + ((size_t)((k >> 1) + half)) * N;
#pragma unroll
            for (int j = 0; j < NT; ++j) {
                const v2f b = bp[(tn0 + j) * 16 + ml];
                acc[j] = __builtin_amdgcn_wmma_f32_16x16x4_f32(
                    false, a, false, b, (short)0, acc[j], false, false);
            }
        }
    }

#pragma unroll
    for (int j = 0; j < NT; ++j) {
        const int n = (tn0 + j) * 16 + ml;
        float bv = 0.f;
        if (bias)  bv += bias[n];
        if (bias2) bv += bias2[n];
#pragma unroll
        for (int r = 0; r < 8; ++r) {
            const int rr = tm * 16 + half * 8 + r;  // ISA C/D layout: vgpr r -> row r / r+8
            if (rr < M) C[(size_t)rr * N + n] = acc[j][r] + bv;
        }
    }
}

// ---------------------------------------------------------------------------
// GCN aggregate: out[g,v,c] = relu(bias[c] + dinv[v]^2*h[g,v,c] + sum_e w_e*h[g,src_e,c])
// blockDim.x == C (channel per lane, coalesced); grid = (1, N, BT).
// ---------------------------------------------------------------------------
__global__ void k_gcn_aggregate(const float* __restrict__ h,
                                const int* __restrict__ off,
                                const int* __restrict__ csrc,
                                const float* __restrict__ cw,
                                const float* __restrict__ dinv,
                                const float* __restrict__ bias,
                                float* __restrict__ out, int C)
{
    const int c = threadIdx.x;
    const int v = blockIdx.y;
    const int g = blockIdx.z;
    const float* hg = h + (size_t)g * 256 * C;
    float s = dinv[v] * dinv[v] * hg[(size_t)v * C + c];
    const int e0 = off[v], e1 = off[v + 1];
    for (int e = e0; e < e1; ++e)
        s += cw[e] * hg[(size_t)csrc[e] * C + c];
    s += bias[c];
    out[((size_t)g * 256 + v) * C + c] = fmaxf(s, 0.f);
}

// pooled[bt,c] = mean over 256 nodes; grid = (BT), block = C
__global__ void k_mean_pool(const float* __restrict__ x, float* __restrict__ pooled, int C)
{
    const int bt = blockIdx.x, c = threadIdx.x;
    float s = 0.f;
    for (int n = 0; n < 256; ++n) s += x[((size_t)bt * 256 + n) * C + c];
    pooled[(size_t)bt * C + c] = s * (1.f / 256.f);
}

// single-step LSTM from zero state: h = sig(o)*tanh(sig(i)*tanh(g))
// gates [BT,1024] gate order i,f,g,o; writes lstm_out[:, dir*256 + c]
__global__ void k_lstm_act(const float* __restrict__ gates, float* __restrict__ out, int dir)
{
    const int r = blockIdx.x, c = threadIdx.x;   // block = 256
    const float* gp = gates + (size_t)r * 1024;
    const float gi = gp[c], gg = gp[512 + c], go = gp[768 + c];
    const float cs = (1.f / (1.f + __expf(-gi))) * tanhf(gg);
    const float hh = (1.f / (1.f + __expf(-go))) * tanhf(cs);
    out[(size_t)r * 512 + dir * 256 + c] = hh;
}

// temporal attention: per-batch softmax over T=32, weighted channel sum.
// grid = (B=8), block = 256 (channel threads; first 32 also compute scores)
__global__ void k_attention(const float* __restrict__ emb, const float* __restrict__ aW,
                            const float* __restrict__ ab, float* __restrict__ xw)
{
    __shared__ float sc[32];
    const int b = blockIdx.x, t = threadIdx.x;
    if (t < 32) {
        const float* e = emb + ((size_t)b * 32 + t) * 256;
        float s = ab[0];
        for (int k = 0; k < 256; ++k) s += e[k] * aW[k];
        sc[t] = s;
    }
    __syncthreads();
    if (t == 0) {
        float m = sc[0];
        for (int i = 1; i < 32; ++i) m = fmaxf(m, sc[i]);
        float den = 0.f;
        for (int i = 0; i < 32; ++i) { sc[i] = __expf(sc[i] - m); den += sc[i]; }
        const float inv = 1.f / den;
        for (int i = 0; i < 32; ++i) sc[i] *= inv;
    }
    __syncthreads();
    float s = 0.f;
    for (int tt = 0; tt < 32; ++tt) s += emb[((size_t)b * 32 + tt) * 256 + t] * sc[tt];
    xw[(size_t)b * 256 + t] = s;
}

// ---------------------------------------------------------------------------
extern "C" void kernel_launch(void* const* d_in, const int* in_sizes, int n_in,
                              void* d_out, int out_size, void* d_ws, size_t ws_size,
                              hipStream_t stream)
{
    (void)n_in; (void)out_size; (void)ws_size;
    const float* data = (const float*)d_in[0];
    const int*   ei   = (const int*)  d_in[1];
    const float* W1 = (const float*)d_in[2];  const float* b1 = (const float*)d_in[3];
    const float* W2 = (const float*)d_in[4];  const float* b2 = (const float*)d_in[5];
    const float* W3 = (const float*)d_in[6];  const float* b3 = (const float*)d_in[7];
    const float* W4 = (const float*)d_in[8];  const float* b4 = (const float*)d_in[9];
    const float* Wih_f = (const float*)d_in[10];
    const float* bih_f = (const float*)d_in[12];
    const float* bhh_f = (const float*)d_in[13];
    const float* Wih_b = (const float*)d_in[14];
    const float* bih_b = (const float*)d_in[16];
    const float* bhh_b = (const float*)d_in[17];
    const float* fcW = (const float*)d_in[18]; const float* fcb = (const float*)d_in[19];
    const float* aW  = (const float*)d_in[20]; const float* ab  = (const float*)d_in[21];
    const float* oW  = (const float*)d_in[22]; const float* ob  = (const float*)d_in[23];
    float* outp = (float*)d_out;

    const int E  = in_sizes[1] / 2;   // 4096
    const int BT = 256;               // B*T = 8*32
    const int R  = BT * 256;          // 65536 node rows

    // workspace layout (floats; all offsets even -> v2f-aligned)
    float* bufA    = (float*)d_ws;                     // [R,256] max
    float* bufB    = bufA    + (size_t)R * 256;        // [R,256] max
    float* pooled  = bufB    + (size_t)R * 256;        // [BT,256]
    float* gates_f = pooled  + (size_t)BT * 256;       // [BT,1024]
    float* gates_b = gates_f + (size_t)BT * 1024;      // [BT,1024]
    float* lstm_o  = gates_b + (size_t)BT * 1024;      // [BT,512]
    float* emb     = lstm_o  + (size_t)BT * 512;       // [BT,256]
    float* xw      = emb     + (size_t)BT * 256;       // [8,256]
    v2f*   Bp      = (v2f*)(xw + 8 * 256);             // [max (K/2)*N = 128*1024] v2f
    float* csr_w   = (float*)(Bp + 131072);            // [E]
    float* dinv    = csr_w + E;                        // [256]
    int*   csr_off = (int*)(dinv + 256);               // [257]
    int*   csr_src = csr_off + 257;                    // [E]

    auto nblk = [](long waves) { return (unsigned)((waves + 7) / 8); };
    auto pblk = [](int K, int N) { return (unsigned)(((K / 2) * N + 255) / 256); };

    k_build_graph<<<1, 256, 0, stream>>>(ei, E, csr_off, csr_src, csr_w, dinv);

    // ---- GCN layer 1: [R,64] @ [64,64] -> aggregate+bias+relu
    k_pack_B<<<pblk(64, 64), 256, 0, stream>>>(W1, Bp, 64, 64, 0);
    k_wmma_gemm<4><<<nblk((R / 16) * 1), 256, 0, stream>>>(data, Bp, nullptr, nullptr, bufB, R, 64, 64);
    k_gcn_aggregate<<<dim3(1, 256, BT), 64, 0, stream>>>(bufB, csr_off, csr_src, csr_w, dinv, b1, bufA, 64);

    // ---- GCN layer 2: [R,64] @ [64,128]
    k_pack_B<<<pblk(64, 128), 256, 0, stream>>>(W2, Bp, 64, 128, 0);
    k_wmma_gemm<8><<<nblk((R / 16) * 1), 256, 0, stream>>>(bufA, Bp, nullptr, nullptr, bufB, R, 128, 64);
    k_gcn_aggregate<<<dim3(1, 256, BT), 128, 0, stream>>>(bufB, csr_off, csr_src, csr_w, dinv, b2, bufA, 128);

    // ---- GCN layer 3: [R,128] @ [128,256]
    k_pack_B<<<pblk(128, 256), 256, 0, stream>>>(W3, Bp, 128, 256, 0);
    k_wmma_gemm<8><<<nblk((R / 16) * 2), 256, 0, stream>>>(bufA, Bp, nullptr, nullptr, bufB, R, 256, 128);
    k_gcn_aggregate<<<dim3(1, 256, BT), 256, 0, stream>>>(bufB, csr_off, csr_src, csr_w, dinv, b3, bufA, 256);

    // ---- GCN layer 4: [R,256] @ [256,256]
    k_pack_B<<<pblk(256, 256), 256, 0, stream>>>(W4, Bp, 256, 256, 0);
    k_wmma_gemm<8><<<nblk((R / 16) * 2), 256, 0, stream>>>(bufA, Bp, nullptr, nullptr, bufB, R, 256, 256);
    k_gcn_aggregate<<<dim3(1, 256, BT), 256, 0, stream>>>(bufB, csr_off, csr_src, csr_w, dinv, b4, bufA, 256);

    // ---- mean pool over nodes -> [BT,256]
    k_mean_pool<<<BT, 256, 0, stream>>>(bufA, pooled, 256);

    // ---- LSTM gates: pooled[BT,256] @ W_ih^T[256,1024] + b_ih + b_hh
    k_pack_B<<<pblk(256, 1024), 256, 0, stream>>>(Wih_f, Bp, 256, 1024, 1);
    k_wmma_gemm<8><<<nblk(16 * 8), 256, 0, stream>>>(pooled, Bp, bih_f, bhh_f, gates_f, BT, 1024, 256);
    k_pack_B<<<pblk(256, 1024), 256, 0, stream>>>(Wih_b, Bp, 256, 1024, 1);
    k_wmma_gemm<8><<<nblk(16 * 8), 256, 0, stream>>>(pooled, Bp, bih_b, bhh_b, gates_b, BT, 1024, 256);
    k_lstm_act<<<BT, 256, 0, stream>>>(gates_f, lstm_o, 0);
    k_lstm_act<<<BT, 256, 0, stream>>>(gates_b, lstm_o, 1);

    // ---- fc: [BT,512] @ [512,256] + fc_b -> emb
    k_pack_B<<<pblk(512, 256), 256, 0, stream>>>(fcW, Bp, 512, 256, 0);
    k_wmma_gemm<8><<<nblk(16 * 2), 256, 0, stream>>>(lstm_o, Bp, fcb, nullptr, emb, BT, 256, 512);

    // ---- temporal attention -> xw [8,256]
    k_attention<<<8, 256, 0, stream>>>(emb, aW, ab, xw);

    // ---- output: [8,256] @ [256,512] + out_b -> d_out [8,512]
    k_pack_B<<<pblk(256, 512), 256, 0, stream>>>(oW, Bp, 256, 512, 0);
    k_wmma_gemm<8><<<nblk(1 * 4), 256, 0, stream>>>(xw, Bp, ob, nullptr, outp, 8, 512, 256);
}